// expiTorch_81097572483400
// MI455X (gfx1250) — compile-verified
//
#include <hip/hip_runtime.h>

#define THREADS 256
#define CFDEPTH 32

typedef float v4f __attribute__((ext_vector_type(4)));

// rcp + one Newton step: ~0.5 ulp reciprocal, 3 VALU ops instead of IEEE div expansion
__device__ __forceinline__ float fast_rcp(float a) {
  float r = __builtin_amdgcn_rcpf(a);
  return r * (2.0f - a * r);
}

__device__ __forceinline__ float ei_f32(float x) {
  const float GAMMA = 0.57721566490153286f;
  if (x >= 40.0f) {
    // Ei(x) ~ e^x/x * sum k!/x^k  (terms shrink fast for x >= 40; 14 terms < fp32 eps)
    float rx = fast_rcp(x);
    float t = 1.0f, s = 1.0f;
#pragma unroll
    for (int k = 1; k <= 14; ++k) { t *= (float)k * rx; s += t; }
    return __expf(x) * rx * s;
  } else if (x <= -6.0f) {
    // Ei(x) = -E1(-x), E1 via backward continued fraction (z >= 6, depth 32 ~ fp32 eps)
    float z = -x;
    float f = z + 2.0f * (float)CFDEPTH + 1.0f;
#pragma unroll
    for (int k = CFDEPTH; k >= 1; --k) {
      f = (z + (float)(2 * k - 1)) - (float)(k * k) * __builtin_amdgcn_rcpf(f);
    }
    return -__expf(-z) * fast_rcp(f);
  } else {
    // Ei(x) = gamma + ln|x| + sum x^n/(n*n!), wave-uniform early exit every 8 terms
    float t = 1.0f, s = 0.0f;
    for (int n = 1; n <= 120; ++n) {
      float fn = (float)n;
      float r = fast_rcp(fn);
      t *= x * r;
      s = fmaf(t, r, s);
      if ((n & 7) == 0 &&
          __builtin_amdgcn_ballot_w32(__builtin_fabsf(t) >
                                      5.9e-8f * __builtin_fabsf(s)) == 0u)
        break;
    }
    return GAMMA + __logf(__builtin_fabsf(x)) + s;
  }
}

__global__ __launch_bounds__(THREADS) void ei_kernel(const float* __restrict__ in,
                                                     float* __restrict__ out,
                                                     long long n) {
  // Double-buffered async staging: each lane owns a private 16B LDS slot per buffer,
  // so no workgroup barriers are needed -- only this wave's ASYNCcnt.
  __shared__ v4f stage[2][THREADS];

  const long long n4     = n >> 2;
  const int       tid    = threadIdx.x;
  const long long gid    = (long long)blockIdx.x * blockDim.x + tid;
  const long long stride = (long long)gridDim.x * blockDim.x;

  const v4f* __restrict__ in4  = (const v4f*)in;
  v4f* __restrict__       out4 = (v4f*)out;

  // Low 32 bits of a flat pointer into LDS == workgroup-relative LDS byte offset
  unsigned lds_addr[2];
  lds_addr[0] = (unsigned)(size_t)&stage[0][tid];
  lds_addr[1] = (unsigned)(size_t)&stage[1][tid];

  int buf = 0;
  if (gid < n4) {
    // input is consumed exactly once -> non-temporal read stream
    asm volatile("global_load_async_to_lds_b128 %0, %1, off th:TH_LOAD_NT"
                 :: "v"(lds_addr[0]), "v"(in4 + gid) : "memory");
  }

  for (long long i = gid; i < n4; i += stride) {
    const long long nxt = i + stride;
    if (nxt < n4) {
      // prefetch next tile into the other buffer, then wait only for the older copy
      asm volatile("global_load_async_to_lds_b128 %0, %1, off th:TH_LOAD_NT"
                   :: "v"(lds_addr[buf ^ 1]), "v"(in4 + nxt) : "memory");
      asm volatile("s_wait_asynccnt 0x1" ::: "memory");
    } else {
      asm volatile("s_wait_asynccnt 0x0" ::: "memory");
    }

    v4f v = stage[buf][tid];
    v4f r;
    r[0] = ei_f32(v[0]);
    r[1] = ei_f32(v[1]);
    r[2] = ei_f32(v[2]);
    r[3] = ei_f32(v[3]);
    // output is written once, never re-read -> non-temporal store stream
    __builtin_nontemporal_store(r, &out4[i]);

    buf ^= 1;
  }

  // scalar tail for n % 4 (no-op for the 8192x4096 case)
  const long long base = n4 << 2;
  if (base + gid < n) out[base + gid] = ei_f32(in[base + gid]);
}

extern "C" void kernel_launch(void* const* d_in, const int* in_sizes, int n_in,
                              void* d_out, int out_size, void* d_ws, size_t ws_size,
                              hipStream_t stream) {
  const float* in  = (const float*)d_in[0];
  float*       out = (float*)d_out;
  long long n  = (long long)in_sizes[0];
  long long n4 = n >> 2;

  int blocks = (int)((n4 + THREADS - 1) / THREADS);
  if (blocks > 2048) blocks = 2048;  // grid-stride: keeps >1 tile per block so the
  if (blocks < 1)    blocks = 1;     // async double-buffer pipeline actually overlaps

  ei_kernel<<<blocks, THREADS, 0, stream>>>(in, out, n);
}